// Enframe_66503273612039
// MI455X (gfx1250) — compile-verified
//
#include <hip/hip_runtime.h>
#include <cstdint>

// Enframe: out[ch*2048 + i, f] = x[ch, i + 512*f]
//   x: (16,4,160000) f32  ->  out: (16, 8192, 309) f32
// Pure data movement: ~41 MB read (L2-resident) + ~162 MB write -> ~8.7 us at 23.3 TB/s.
// gfx1250 path: async global->LDS copy (ASYNCcnt) + skewed-LDS transpose for
// fully coalesced, conflict-free stores along the innermost frame dimension.

#define S_LEN    160000
#define FRAME    2048
#define HOP      512
#define NFRAMES  309            // (160000-2048)/512 + 1
#define FT_MAX   16             // frames per tile
#define NTILES   20             // ceil(309/16)
#define THREADS  256
#define LDS_BYTES 39232         // 9728 samples*4B + 4-word pad per 512-sample segment (9800 words), rounded up

__global__ __launch_bounds__(THREADS)
void enframe_async_kernel(const float* __restrict__ x, float* __restrict__ out) {
    extern __shared__ float smem[];   // dynamic LDS: starts at LDS offset 0 (no static LDS)

    const uint32_t tile = blockIdx.x;           // 0..NTILES-1
    const uint32_t ch   = blockIdx.y;           // 0..B*C-1
    const uint32_t t    = threadIdx.x;

    const uint32_t f0 = tile * FT_MAX;
    const uint32_t ft = (NFRAMES - f0 < FT_MAX) ? (NFRAMES - f0) : FT_MAX;
    // contiguous input samples needed by this tile: [512*f0, 512*f0 + 512*(ft-1) + 2048)
    const uint32_t chunk_bytes = (HOP * (ft - 1u) + FRAME) * 4u;   // multiple of 16

    // ---------------- async global -> LDS copy (CDNA5 ASYNCcnt path) ----------------
    // LDS layout (skewed for conflict-free transpose reads, 16B-aligned for B128):
    //   sample s  ->  LDS word  s + 4*(s>>9)   (i.e. +16B pad per 2048B segment)
    const uint64_t gbase =
        (uint64_t)(uintptr_t)(x + (size_t)ch * S_LEN + (size_t)f0 * HOP);
    for (uint32_t off = t * 16u; off < chunk_bytes; off += THREADS * 16u) {
        uint32_t lds_byte = off + ((off >> 11) << 4);   // byte skew, stays 16B aligned
        asm volatile("global_load_async_to_lds_b128 %0, %1, %2"
                     :
                     : "v"(lds_byte), "v"(off), "s"(gbase)
                     : "memory");
    }
    asm volatile("s_wait_asynccnt 0x0" ::: "memory");
    __syncthreads();

    // ---------------- LDS transpose -> coalesced global stores ----------------
    // lane j = frame-in-tile (0..15), 16 rows per pass:
    //   lanes 0-15 write 64 contiguous bytes of row i, lanes 16-31 of row i+1.
    // LDS bank = (i + 4j + 4*(i>>9)) & 63 -> 32 distinct banks per wave.
    const uint32_t j  = t & 15u;
    const uint32_t r0 = t >> 4u;
    float* __restrict__ outr = out + (size_t)ch * FRAME * NFRAMES + f0;
    if (j < ft) {
        #pragma unroll 4
        for (uint32_t p = 0; p < FRAME / 16u; ++p) {
            uint32_t i = p * 16u + r0;
            // sample s = i + 512*j  ->  word s + 4*(s>>9) = i + 516*j + 4*(i>>9)
            float v = smem[i + 516u * j + 4u * (i >> 9u)];
            outr[(size_t)i * NFRAMES + j] = v;
        }
    }
}

extern "C" void kernel_launch(void* const* d_in, const int* in_sizes, int n_in,
                              void* d_out, int out_size, void* d_ws, size_t ws_size,
                              hipStream_t stream) {
    (void)n_in; (void)out_size; (void)d_ws; (void)ws_size;
    const float* x = (const float*)d_in[0];
    float* out = (float*)d_out;

    const int channels = in_sizes[0] / S_LEN;   // B*C = 64
    dim3 grid(NTILES, channels, 1);
    dim3 block(THREADS, 1, 1);
    enframe_async_kernel<<<grid, block, LDS_BYTES, stream>>>(x, out);
}